// FullTensorProductSparse_90263032693452
// MI455X (gfx1250) — compile-verified
//
#include <hip/hip_runtime.h>

typedef __attribute__((ext_vector_type(2))) float v2f;
typedef __attribute__((ext_vector_type(8))) float v8f;
typedef int v4i __attribute__((vector_size(16)));

#define NLEAD 2048
#define OUTW  32768
#define LDS_BYTES (512 * 44 * 4)

#if __has_builtin(__builtin_amdgcn_global_store_async_from_lds_b128) && \
    __has_builtin(__builtin_amdgcn_s_wait_asynccnt)
#define USE_ASYNC_FLUSH 1
#else
#define USE_ASYNC_FLUSH 0
#endif

// One rank<=4 plane: D(16x16) = A(16x4) @ B(4x16), scattered to LDS.
// A layout (ISA 7.12.2, 32-bit A 16x4): vgpr0 = K0 (lanes 0-15) / K2 (lanes 16-31),
// vgpr1 = K1 / K3.  B mirrors (rows striped across lanes).  Caller passes the
// half-selected per-lane values A0/A1/B0/B1.
__device__ __forceinline__ void wmma_to_lds(float* lds, float A0, float A1,
                                            float B0, float B1,
                                            int ldsbase, int rstride) {
  v2f A; A[0] = A0; A[1] = A1;
  v2f B; B[0] = B0; B[1] = B1;
  v8f C = {0.f, 0.f, 0.f, 0.f, 0.f, 0.f, 0.f, 0.f};
  v8f D = __builtin_amdgcn_wmma_f32_16x16x4_f32(false, A, false, B, (short)0, C,
                                                false, false);
#pragma unroll
  for (int r = 0; r < 8; ++r)
    lds[ldsbase + r * rstride] = D[r];  // D vgpr r: M=r (lo half), M=8+r (hi half)
}

// Flush one segment: 512 local s-rows x (8*D3) contiguous output columns.
// Fully unrolled (trip counts are exact multiples of the 256-thread block).
template <int D3, int SEGBASE>
__device__ __forceinline__ void flush_seg(float* lds, float* __restrict__ out,
                                          int tid, int ut, int sb) {
  constexpr int W     = 8 * D3;   // contiguous floats per row (sigma*d3+k runs)
  constexpr int P     = W + 4;    // LDS pitch (dwords), mult of 4, bank-friendly
  constexpr int NF4   = W / 4;
  constexpr int ITERS = (512 * NF4) / 256;
#pragma unroll
  for (int it = 0; it < ITERS; ++it) {
    const int idx = tid + it * 256;
    const int row = idx / NF4;
    const int c4  = idx - row * NF4;
    const size_t o =
        (size_t)(ut * 512 + row) * OUTW + SEGBASE + (size_t)sb * W + c4 * 4;
#if USE_ASYNC_FLUSH
    // LDS -> global DMA, 16B per lane, tracked on ASYNCcnt (no VGPR staging).
    float* gp = out + o;
    float* lp = lds + row * P + c4 * 4;
    __attribute__((address_space(1))) float* gp1 =
        (__attribute__((address_space(1))) float*)gp;
    __attribute__((address_space(3))) float* lp3 =
        (__attribute__((address_space(3))) float*)lp;
    __builtin_amdgcn_global_store_async_from_lds_b128(
        (__attribute__((address_space(1))) v4i*)gp1,
        (__attribute__((address_space(3))) v4i*)lp3, 0, 0);
#else
    float4 val = *reinterpret_cast<const float4*>(lds + row * P + c4 * 4);
    *reinterpret_cast<float4*>(out + o) = val;  // global_store_b128
#endif
  }
#if USE_ASYNC_FLUSH
  // Drain this wave's async stores before signaling the barrier, so no wave
  // overwrites LDS while another wave's async engine is still reading it.
  __builtin_amdgcn_s_wait_asynccnt(0);
#endif
}

__global__ void __launch_bounds__(256)
tp_sparse_kernel(const float* __restrict__ in1, const float* __restrict__ in2,
                 float* __restrict__ out) {
  extern __shared__ float lds[];
  const int tid  = threadIdx.x;
  const int lane = tid & 31;
  const int w    = tid >> 5;
  const int vt   = w & 1;        // v-tile (2 x 16)
  const int ss   = w >> 1;       // sigma subgroup 0..3 (2 sigma each)
  const int half = lane >> 4;    // 0: K0/K1 cols, 1: K2/K3 cols
  const int m    = lane & 15;    // M for A-side, N for B-side
  const int ut   = blockIdx.x & 3;    // u-tile (4 x 16)
  const int sb   = blockIdx.x >> 2;   // sigma batch 0..255 (8 sigma)

  const int u  = ut * 16 + m;
  const int v  = vt * 16 + m;
  const int v0 = vt * 16;

  // Gather the 8 input values per (lane, sigma).  Inputs are L2-resident (3MB).
  float va0[2], vb0[2], va[2][3], vb[2][3];
#pragma unroll
  for (int s2 = 0; s2 < 2; ++s2) {
    const unsigned sg = (unsigned)(sb * 8 + ss * 2 + s2);
    { unsigned q = (unsigned)u * NLEAD + sg; va0[s2] = in1[(q >> 6) * 256 + (q & 63u)]; }
    { unsigned q = (unsigned)v * NLEAD + sg; vb0[s2] = in2[(q >> 5) * 128 + (q & 31u)]; }
#pragma unroll
    for (int i = 0; i < 3; ++i) {
      unsigned q1 = (unsigned)(i * 64 + u) * NLEAD + sg;
      va[s2][i] = in1[(q1 / 192u) * 256 + 64 + (q1 % 192u)];
      unsigned q2 = (unsigned)(i * 32 + v) * NLEAD + sg;
      vb[s2][i] = in2[(q2 / 96u) * 128 + 32 + (q2 % 96u)];
    }
  }

  const float R2 = 0.70710678118654752f;  // 1/sqrt(2)
  const float R3 = 0.57735026918962576f;  // 1/sqrt(3)
  const float R6 = 0.40824829046386302f;  // 1/sqrt(6)

  const int brow = half * 256 + v0 + m;   // lds base row = (half*8)*32 + v

  // ---------- seg 0 @0 (0e x 0e -> 0e): a0*b0, d3=1
  {
    const int P = 12;
#pragma unroll
    for (int s2 = 0; s2 < 2; ++s2) {
      const int sl = ss * 2 + s2;
      wmma_to_lds(lds, half ? 0.f : va0[s2], 0.f,
                       half ? 0.f : vb0[s2], 0.f, brow * P + sl, 32 * P);
    }
    __syncthreads();
    flush_seg<1, 0>(lds, out, tid, ut, sb);
    __syncthreads();
  }
  // ---------- seg 1 @2048 (1o x 1o -> 0e): (a.b)/sqrt3, d3=1, K=3
  {
    const int P = 12;
#pragma unroll
    for (int s2 = 0; s2 < 2; ++s2) {
      const int sl = ss * 2 + s2;
      wmma_to_lds(lds, half ? va[s2][2] : va[s2][0], half ? 0.f : va[s2][1],
                       (half ? vb[s2][2] : vb[s2][0]) * R3,
                       half ? 0.f : vb[s2][1] * R3, brow * P + sl, 32 * P);
    }
    __syncthreads();
    flush_seg<1, 2048>(lds, out, tid, ut, sb);
    __syncthreads();
  }
  // ---------- seg 2 @4096 (0e x 1o -> 1o): a0*b1[k], d3=3
  {
    const int P = 28;
#pragma unroll
    for (int s2 = 0; s2 < 2; ++s2) {
      const int sl = ss * 2 + s2;
#pragma unroll
      for (int k = 0; k < 3; ++k)
        wmma_to_lds(lds, half ? 0.f : va0[s2], 0.f,
                         half ? 0.f : vb[s2][k], 0.f,
                         brow * P + sl * 3 + k, 32 * P);
    }
    __syncthreads();
    flush_seg<3, 4096>(lds, out, tid, ut, sb);
    __syncthreads();
  }
  // ---------- seg 3 @10240 (1o x 0e -> 1o): a1[k]*b0, d3=3
  {
    const int P = 28;
#pragma unroll
    for (int s2 = 0; s2 < 2; ++s2) {
      const int sl = ss * 2 + s2;
#pragma unroll
      for (int k = 0; k < 3; ++k)
        wmma_to_lds(lds, half ? 0.f : va[s2][k], 0.f,
                         half ? 0.f : vb0[s2], 0.f,
                         brow * P + sl * 3 + k, 32 * P);
    }
    __syncthreads();
    flush_seg<3, 10240>(lds, out, tid, ut, sb);
    __syncthreads();
  }
  // ---------- seg 4 @16384 (1o x 1o -> 1e): cross(a,b)/sqrt2, d3=3, K=2
  {
    const int P = 28;
#pragma unroll
    for (int s2 = 0; s2 < 2; ++s2) {
      const int sl = ss * 2 + s2;
#pragma unroll
      for (int k = 0; k < 3; ++k) {
        const int i1 = (k + 1) % 3, i2 = (k + 2) % 3;
        wmma_to_lds(lds, half ? 0.f : va[s2][i1], half ? 0.f : va[s2][i2],
                         half ? 0.f :  vb[s2][i2] * R2,
                         half ? 0.f : -vb[s2][i1] * R2,
                         brow * P + sl * 3 + k, 32 * P);
      }
    }
    __syncthreads();
    flush_seg<3, 16384>(lds, out, tid, ut, sb);
    __syncthreads();
  }
  // ---------- seg 5 @22528 (1o x 1o -> 2e): sym-traceless, d3=5
  {
    const int P = 44;
#pragma unroll
    for (int s2 = 0; s2 < 2; ++s2) {
      const int sl = ss * 2 + s2;
      const float* a = va[s2];
      const float* b = vb[s2];
      // basis order 0=y,1=z,2=x ; m=-2..+2 -> xy, yz, z2, zx, x2-y2
      wmma_to_lds(lds, half ? 0.f : a[2], half ? 0.f : a[0],
                       half ? 0.f : b[0] * R2, half ? 0.f : b[2] * R2,
                       brow * P + sl * 5 + 0, 32 * P);            // (ax*by+ay*bx)/r2
      wmma_to_lds(lds, half ? 0.f : a[0], half ? 0.f : a[1],
                       half ? 0.f : b[1] * R2, half ? 0.f : b[0] * R2,
                       brow * P + sl * 5 + 1, 32 * P);            // (ay*bz+az*by)/r2
      wmma_to_lds(lds, half ? a[0] : a[1], half ? 0.f : a[2],
                       half ? (-b[0] * R6) : (2.f * b[1] * R6),
                       half ? 0.f : (-b[2] * R6),
                       brow * P + sl * 5 + 2, 32 * P);            // (2az*bz-ax*bx-ay*by)/r6
      wmma_to_lds(lds, half ? 0.f : a[1], half ? 0.f : a[2],
                       half ? 0.f : b[2] * R2, half ? 0.f : b[1] * R2,
                       brow * P + sl * 5 + 3, 32 * P);            // (az*bx+ax*bz)/r2
      wmma_to_lds(lds, half ? 0.f : a[2], half ? 0.f : a[0],
                       half ? 0.f : b[2] * R2, half ? 0.f : (-b[0] * R2),
                       brow * P + sl * 5 + 4, 32 * P);            // (ax*bx-ay*by)/r2
    }
    __syncthreads();
    flush_seg<5, 22528>(lds, out, tid, ut, sb);
    __syncthreads();
  }
}

extern "C" void kernel_launch(void* const* d_in, const int* in_sizes, int n_in,
                              void* d_out, int out_size, void* d_ws, size_t ws_size,
                              hipStream_t stream) {
  const float* in1 = (const float*)d_in[0];  // (2048, 256) f32
  const float* in2 = (const float*)d_in[1];  // (2048, 128) f32
  float* out = (float*)d_out;                // (2048, 32768) f32

  (void)in_sizes; (void)n_in; (void)out_size; (void)d_ws; (void)ws_size;

  // 4 u-tiles x 256 sigma-batches; 8 waves/block (wave32).
  (void)hipFuncSetAttribute((const void*)tp_sparse_kernel,
                            hipFuncAttributeMaxDynamicSharedMemorySize, LDS_BYTES);
  tp_sparse_kernel<<<dim3(1024), dim3(256), LDS_BYTES, stream>>>(in1, in2, out);
}